// Conv2dKan_24137716203761
// MI455X (gfx1250) — compile-verified
//
#include <hip/hip_runtime.h>

// ---------------------------------------------------------------------------
// KAN conv (C_IN=64, C_OUT=128, 3x3, pad1) as an fp32 WMMA GEMM on gfx1250.
//   out[b,o,y,x] = sum_{c,tap,f} feat_f(x[b,c,y+kh-1,x+kw-1]) * W[o, c*9+tap, f]
//   feat_0 = silu(u), feat_{1+n} = H_n(u) (physicists' Hermite, n=0..7)
//   W[.,.,0] = w_b ; W[.,.,1+n] = c_n * w_s
// M = 18432 pixels, N = 128 channels, K = 5184 -> 24.5 GFLOP, ~17 MB traffic:
// compute-bound; Hermite dynamic range (~1e6) demands fp32 -> V_WMMA_F32_16X16X4_F32.
//
// v2: fully unroll the 21 K-chunks (compile-time FF/W offsets, no /9 in the
//     hot loop) and pack weights K-pairwise so each lane's B fragment is one
//     contiguous 8-byte LDS read (ds_load_b64).
// ---------------------------------------------------------------------------

typedef __attribute__((ext_vector_type(2))) float v2f;
typedef __attribute__((ext_vector_type(8))) float v8f;

#define CIN     64
#define COUT    128
#define HW      48
#define CKK     576          // 64*9
#define KEL     84           // 9 taps * 9 feats = 81, padded to 84 (21 chunks of 4)
#define NPAIR   (KEL / 2)    // 42 K-pairs
#define TILE_W  16
#define TILE_H  8
#define PATCH_W 18
#define PATCH_H 10
#define NPATCH  (PATCH_W * PATCH_H)   // 180

// ---------------------------------------------------------------------------
// Kernel 1: pack weights into Wp[c][pair][o][2] with kel = 2*pair + j,
// kel = tap*9 + f (f=0 -> w_b, f=1+n -> c_n*w_s), zeros for kel >= 81.
// A lane's two B elements (k = e0, e0+1) are 8 contiguous bytes.
// ---------------------------------------------------------------------------
__global__ void pack_weights_kernel(const float* __restrict__ wb,
                                    const float* __restrict__ wsc,
                                    const float* __restrict__ cs,
                                    float* __restrict__ Wp) {
    int idx = blockIdx.x * blockDim.x + threadIdx.x;
    if (idx >= CIN * KEL * COUT) return;
    int c = idx / (KEL * COUT);
    int r = idx - c * (KEL * COUT);
    int p = r / (COUT * 2);              // K-pair index, 0..41
    int q = r - p * (COUT * 2);
    int o = q >> 1;
    int j = q & 1;
    int kel = 2 * p + j;
    float val = 0.0f;
    if (kel < 81) {
        int tap = kel / 9;
        int f   = kel - tap * 9;
        int k   = c * 9 + tap;           // channel-major (c,kh,kw) = unfold order
        if (f == 0) val = wb[o * CKK + k];
        else        val = cs[((f - 1) * COUT + o) * CKK + k] * wsc[o * CKK + k];
    }
    Wp[idx] = val;
}

// A-fragment element: feature value for pixel column m of wave's image row,
// flat K element e (compile-time constant after unrolling).
__device__ __forceinline__ float featA(const float* __restrict__ FF,
                                       int wave, int m, int e) {
    int tap = e / 9;
    int f   = e - tap * 9;
    int kh  = tap / 3;
    int kw  = tap - kh * 3;
    float v = FF[((wave + kh) * PATCH_W + (m + kw)) * 9 + f];
    return (e < 81) ? v : 0.0f;          // padded K lanes contribute 0
}

// ---------------------------------------------------------------------------
// Kernel 2: fused feature + GEMM. One WG = 16x8 pixel tile x 128 channels.
// 8 waves; wave w owns image row y0+w (16 pixels = one WMMA M-tile) and all
// 8 o-tiles (64 accumulator VGPRs). Unfold is implicit via shifted LDS reads.
// ---------------------------------------------------------------------------
__global__ void __launch_bounds__(256)
kan_conv_wmma_kernel(const float* __restrict__ xg,
                     const float* __restrict__ Wp,
                     float* __restrict__ out) {
    __shared__ __align__(16) float FF[NPATCH * 9];   //  6480 B: 9 feats per patch value
    __shared__ __align__(16) float Wl[KEL * COUT];   // 43008 B: [pair][o][2] weight slice

    const int tid  = threadIdx.x;
    const int wave = tid >> 5;           // 0..7 (wave32)
    const int lane = tid & 31;
    const int half = lane >> 4;          // K/M split per ISA §7.12.2
    const int n16  = lane & 15;

    const int blk = blockIdx.x;          // 8 batches * 6 y-tiles * 3 x-tiles = 144
    const int b   = blk / 18;
    const int rem = blk - b * 18;
    const int ty  = rem / 3;
    const int tx  = rem - ty * 3;
    const int y0  = ty * TILE_H;
    const int x0  = tx * TILE_W;

    v8f acc[8] = {};                     // 8 o-tiles of 16x16 fp32

    const int py = tid / PATCH_W;        // patch loader coords (threads 0..179)
    const int px = tid - py * PATCH_W;

    const float2* Wl2 = (const float2*)Wl;   // [pair][o] as float2

    for (int c = 0; c < CIN; ++c) {
        __syncthreads();                 // protect LDS from previous iteration's readers

        // ---- stage weight slice (42x128 float2) via float4 copies ----
        {
            const float4* src = (const float4*)(Wp + c * (KEL * COUT));
            float4*       dst = (float4*)Wl;
            #pragma unroll
            for (int j = 0; j < (KEL * COUT) / 4 / 256 + 1; ++j) {
                int i = tid + j * 256;
                if (i < (KEL * COUT) / 4) dst[i] = src[i];
            }
        }
        if (c + 1 < CIN)                 // gfx1250 global_prefetch_b8 into L2
            __builtin_prefetch(Wp + (c + 1) * (KEL * COUT), 0, 1);

        // ---- halo patch load + feature evaluation (once per x value) ----
        if (tid < NPATCH) {
            int gy = y0 + py - 1;
            int gx = x0 + px - 1;
            bool inb = (gy >= 0) & (gy < HW) & (gx >= 0) & (gx < HW);
            float v = inb ? xg[((b * CIN + c) * HW + gy) * HW + gx] : 0.0f;
            float* ff = FF + tid * 9;
            ff[0] = v * (1.0f / (1.0f + __expf(-v)));          // silu
            float hm2 = 1.0f;                                   // H_0
            float hm1 = 2.0f * v;                               // H_1
            ff[1] = hm2;
            ff[2] = hm1;
            #pragma unroll
            for (int i = 1; i <= 6; ++i) {                      // H_2..H_7
                float h = 2.0f * v * hm1 - 2.0f * (float)i * hm2;
                ff[2 + i] = h;
                hm2 = hm1; hm1 = h;
            }
        }
        __syncthreads();

        // ---- 21 K-chunks of 4, 8 N-tiles each: V_WMMA_F32_16X16X4_F32 ----
        // Fully unrolled: all FF/Wl offsets are compile-time constants.
        #pragma unroll
        for (int cb = 0; cb < 21; ++cb) {
            const int e0 = cb * 4 + half * 2;    // this lane's 2 K elements
            v2f a;
            a.x = featA(FF, wave, n16, e0);
            a.y = featA(FF, wave, n16, e0 + 1);
            const float2* w0 = Wl2 + (cb * 2 + half) * COUT + n16;
            #pragma unroll
            for (int t = 0; t < 8; ++t) {
                float2 wv = w0[t * 16];          // one ds_load_b64: B[e0][n], B[e0+1][n]
                v2f bf;
                bf.x = wv.x;
                bf.y = wv.y;
                acc[t] = __builtin_amdgcn_wmma_f32_16x16x4_f32(
                    /*neg_a=*/false, a, /*neg_b=*/false, bf,
                    /*c_mod=*/(short)0, acc[t],
                    /*reuse_a=*/false, /*reuse_b=*/false);
            }
        }
    }

    // ---- store: D layout => lane writes 8 consecutive x's per o-tile ----
    const int y = y0 + wave;
    #pragma unroll
    for (int t = 0; t < 8; ++t) {
        const int o = t * 16 + n16;
        float* p = out + ((b * COUT + o) * HW + y) * HW + x0 + half * 8;
        float4 lo, hi;
        lo.x = acc[t][0]; lo.y = acc[t][1]; lo.z = acc[t][2]; lo.w = acc[t][3];
        hi.x = acc[t][4]; hi.y = acc[t][5]; hi.z = acc[t][6]; hi.w = acc[t][7];
        ((float4*)p)[0] = lo;
        ((float4*)p)[1] = hi;
    }
}

// ---------------------------------------------------------------------------
extern "C" void kernel_launch(void* const* d_in, const int* in_sizes, int n_in,
                              void* d_out, int out_size, void* d_ws, size_t ws_size,
                              hipStream_t stream) {
    const float* x   = (const float*)d_in[0];   // (8,64,48,48)
    const float* wb  = (const float*)d_in[1];   // (128,576,1)
    const float* wsc = (const float*)d_in[2];   // (128,576,1)
    const float* cs  = (const float*)d_in[3];   // (8,128,576,1)
    float*       o   = (float*)d_out;           // (8,128,48,48)
    float*       Wp  = (float*)d_ws;            // 64*84*128 floats = 2.75 MB

    const int wtot = CIN * KEL * COUT;
    pack_weights_kernel<<<(wtot + 255) / 256, 256, 0, stream>>>(wb, wsc, cs, Wp);
    kan_conv_wmma_kernel<<<dim3(8 * 6 * 3), dim3(256), 0, stream>>>(x, Wp, o);
}